// ARGCN_39805756900152
// MI455X (gfx1250) — compile-verified
//
#include <hip/hip_runtime.h>
#include <hip/hip_bf16.h>

// ---------------------------------------------------------------------------
// ARGCN forward for MI455X (gfx1250, wave32, WMMA).
// All matrix math uses V_WMMA_F32_16X16X4_F32 (full f32 — workload is
// bandwidth-bound at 23.3 TB/s, so there is no reason to drop precision).
// B operands are staged in LDS with the two K-values of each WMMA fragment
// interleaved so the fragment is a single aligned ds_load_b64 (no v_mov
// repacking in the inner loop).
// ---------------------------------------------------------------------------

typedef float v2f __attribute__((ext_vector_type(2)));
typedef float v8f __attribute__((ext_vector_type(8)));

#define N_NODES 50000
#define N_EDGES 500000
#define DMODEL  256
#define NH      4
#define DV      32
#define HKV     128      // NH*DV
#define ATTD    32
#define RPD     32
#define CATK    96       // 2*DV + RPD  (also DRD+DAD+ATT for edge_w)
#define MAXRP   64
#define EPSV    1e-12f

#define ETILES  (N_EDGES/16)   // 31250
#define MTILES  (N_NODES/16)   // 3125

__device__ __forceinline__ float lrelu(float x) { return x >= 0.f ? x : 0.2f * x; }

// order-preserving float<->uint mapping for atomicMax over signed floats
__device__ __forceinline__ unsigned fenc(float f) {
    unsigned u = __float_as_uint(f);
    return (u & 0x80000000u) ? ~u : (u | 0x80000000u);
}
__device__ __forceinline__ float fdec(unsigned k) {
    return __uint_as_float((k & 0x80000000u) ? (k & 0x7FFFFFFFu) : ~k);
}

__device__ __forceinline__ v8f vzero8() {
    v8f z = {0.f, 0.f, 0.f, 0.f, 0.f, 0.f, 0.f, 0.f};
    return z;
}

// D = A(16x4 f32) * B(4x16 f32) + C(16x16 f32)
__device__ __forceinline__ v8f wmma_f32(v2f a, v2f b, v8f c) {
    return __builtin_amdgcn_wmma_f32_16x16x4_f32(false, a, false, b, (short)0, c,
                                                 false, false);
}

// ---------------------------------------------------------------------------
// Generic row-major WMMA GEMM: C[M,Nn] = A[M,K] * B[K,Nn] (+bias[Nn]).
// Requires K % 128 == 0 (true for K=256 and K=128 here).
// Block = 256 threads (8 waves). Each wave owns one 16-row M tile; the whole
// block shares one 64-column N panel of B staged through LDS in 128-deep
// K chunks. A is read directly (float2, coalesced over 16 rows).
// A layout (16x4 f32): lane&15 = row M, lane>>4 selects K pair {0,1}/{2,3}.
// B layout (4x16 f32): lane&15 = col N, lane>>4 selects K pair; b.x=K, b.y=K+1.
// LDS B layout is pair-interleaved: lds_b[((k>>1)*64 + col)*2 + (k&1)] so a
// B fragment is one aligned 8-byte ds_load_b64.
// C layout: VGPR j -> row j (lanes 0-15) / row j+8 (lanes 16-31), col = lane&15.
// ---------------------------------------------------------------------------
__global__ __launch_bounds__(256) void gemm_wmma(const float* __restrict__ A,
                                                 const float* __restrict__ B,
                                                 const float* __restrict__ bias,
                                                 float* __restrict__ C,
                                                 int M, int Nn, int K) {
    __shared__ float lds_b[128 * 64];   // 32 KB K-chunk x 64-col panel

    const int tid  = threadIdx.x;
    const int wave = tid >> 5;
    const int lane = tid & 31;
    const int ln   = lane & 15;
    const int hi   = lane >> 4;

    const int mt = blockIdx.x * 8 + wave;
    const int m0 = mt * 16;
    const int n0 = blockIdx.y * 64;
    const bool active = (m0 < M);

    v8f acc[4];
#pragma unroll
    for (int t = 0; t < 4; ++t) acc[t] = vzero8();

    for (int kc = 0; kc < K; kc += 128) {
        // cooperative stage of B[kc..kc+128, n0..n0+64], pair-interleaved
        for (int idx = tid; idx < 128 * 64; idx += 256) {
            const int kk  = idx >> 6;       // 0..127
            const int col = idx & 63;
            lds_b[(((kk >> 1) * 64) + col) * 2 + (kk & 1)] =
                B[(size_t)(kc + kk) * Nn + n0 + col];
        }
        __syncthreads();

        if (active) {
            if (kc + 128 < K)  // prefetch next A chunk (global_prefetch_b8)
                __builtin_prefetch(&A[(size_t)(m0 + ln) * K + kc + 128], 0, 0);
#pragma unroll 4
            for (int k0 = 0; k0 < 128; k0 += 4) {
                const int kb = k0 + hi * 2;
                v2f a = *reinterpret_cast<const v2f*>(
                    &A[(size_t)(m0 + ln) * K + kc + kb]);
#pragma unroll
                for (int t = 0; t < 4; ++t) {
                    v2f b = *reinterpret_cast<const v2f*>(
                        &lds_b[(((kb >> 1) * 64) + t * 16 + ln) * 2]);
                    acc[t] = wmma_f32(a, b, acc[t]);
                }
            }
        }
        __syncthreads();
    }

    if (active) {
#pragma unroll
        for (int t = 0; t < 4; ++t) {
#pragma unroll
            for (int j = 0; j < 8; ++j) {
                int row = m0 + j + hi * 8;
                int col = n0 + t * 16 + ln;
                float val = acc[t][j];
                if (bias) val += bias[col];
                C[(size_t)row * Nn + col] = val;
            }
        }
    }
}

// ---------------------------------------------------------------------------
// Edge attention scores: for each head h and each 16-edge tile, build the
// att_cat tile [16 x 96] = [kq[src] | kq[tgt] | E_pos] in LDS, multiply by
// att_w[h] [96 x 32] with WMMA, apply leaky-relu, store scores [E,H,32] and
// fold a block-level max into the global max scratch.
// att_w staged pair-interleaved like the GEMM B panel.
// ---------------------------------------------------------------------------
__global__ __launch_bounds__(256) void edge_att_kernel(
    const float* __restrict__ kq, const int* __restrict__ src,
    const int* __restrict__ tgt, const int* __restrict__ relpos,
    const float* __restrict__ Epos, const float* __restrict__ att_w,
    float* __restrict__ scores, unsigned* __restrict__ maxscr) {
    __shared__ float lds_aw[CATK * ATTD];     // 12 KB  att_w[h], pair-interleaved
    __shared__ float lds_cat[8][16 * CATK];   // 48 KB  per-wave cat tiles
    __shared__ float red[256];

    const int tid  = threadIdx.x;
    const int wave = tid >> 5;
    const int lane = tid & 31;
    const int ln   = lane & 15;
    const int hi   = lane >> 4;
    const int h    = blockIdx.y;
    const int gt   = blockIdx.x * 8 + wave;
    const bool active = gt < ETILES;

    for (int idx = tid; idx < CATK * ATTD; idx += 256) {
        const int kk = idx >> 5;   // 0..95
        const int nn = idx & 31;
        lds_aw[(((kk >> 1) * ATTD) + nn) * 2 + (kk & 1)] =
            att_w[(size_t)h * CATK * ATTD + idx];
    }

    if (active) {
        const int e0 = gt * 16;
        float* cat = lds_cat[wave];
        for (int e = 0; e < 16; ++e) {
            const int eg = e0 + e;
            const int s  = src[eg];
            const int t  = tgt[eg];
            const int pi = relpos[eg] + MAXRP;
            const int c  = lane;  // 32 lanes cover 3 x 32 columns
            cat[e * CATK + c]      = kq[(size_t)s * HKV + h * DV + c];
            cat[e * CATK + 32 + c] = kq[(size_t)t * HKV + h * DV + c];
            cat[e * CATK + 64 + c] = Epos[(size_t)pi * RPD + c];
        }
    }
    __syncthreads();

    float lmax = -__builtin_inff();
    if (active) {
        const float* cat = lds_cat[wave];
        v8f acc0 = vzero8(), acc1 = vzero8();
#pragma unroll 4
        for (int k0 = 0; k0 < CATK; k0 += 4) {
            const int kb = k0 + hi * 2;
            v2f a = *reinterpret_cast<const v2f*>(&cat[ln * CATK + kb]);
            v2f b0 = *reinterpret_cast<const v2f*>(
                &lds_aw[(((kb >> 1) * ATTD) + ln) * 2]);
            v2f b1 = *reinterpret_cast<const v2f*>(
                &lds_aw[(((kb >> 1) * ATTD) + 16 + ln) * 2]);
            acc0 = wmma_f32(a, b0, acc0);
            acc1 = wmma_f32(a, b1, acc1);
        }
        const int e0 = gt * 16;
#pragma unroll
        for (int j = 0; j < 8; ++j) {
            const int row = e0 + j + hi * 8;
            const float s0 = lrelu(acc0[j]);
            const float s1 = lrelu(acc1[j]);
            scores[((size_t)row * NH + h) * ATTD + ln]      = s0;
            scores[((size_t)row * NH + h) * ATTD + 16 + ln] = s1;
            lmax = fmaxf(lmax, fmaxf(s0, s1));
        }
    }

    red[tid] = lmax;
    __syncthreads();
    for (int s = 128; s > 0; s >>= 1) {
        if (tid < s) red[tid] = fmaxf(red[tid], red[tid + s]);
        __syncthreads();
    }
    if (tid == 0) atomicMax(maxscr, fenc(red[0]));
}

// exp(score - globalmax), stored in place; atomic segment-sum into denom[tgt]
__global__ __launch_bounds__(256) void exp_denom_kernel(
    float* __restrict__ scores, const int* __restrict__ tgt,
    const unsigned* __restrict__ maxscr, float* __restrict__ denom) {
    const size_t idx = (size_t)blockIdx.x * 256 + threadIdx.x;
    if (idx >= (size_t)N_EDGES * HKV) return;
    const float M = fdec(*maxscr);
    const int e   = (int)(idx >> 7);
    const int rem = (int)(idx & 127);
    const float v = __expf(scores[idx] - M);
    scores[idx] = v;
    atomicAdd(&denom[(size_t)tgt[e] * HKV + rem], v);
}

// beta -> edge score (96-dot with edge_w[h]) -> attn output + weighted scatter
__global__ __launch_bounds__(256) void edge_final_kernel(
    const float* __restrict__ expatt, const float* __restrict__ denom,
    const int* __restrict__ src, const int* __restrict__ tgt,
    const int* __restrict__ deprel, const int* __restrict__ deparc,
    const float* __restrict__ Edeprel, const float* __restrict__ Edeparc,
    const float* __restrict__ edge_w, const float* __restrict__ v,
    float* __restrict__ attn_out, float* __restrict__ accum) {
    const int gid = blockIdx.x * 256 + threadIdx.x;
    if (gid >= N_EDGES * NH) return;
    const int e = gid >> 2;
    const int h = gid & 3;
    const int s = src[e];
    const int t = tgt[e];
    const int dr = deprel[e];
    const int da = deparc[e];
    const float* ew = &edge_w[h * CATK];

    float sum = 0.f;
#pragma unroll 8
    for (int j = 0; j < 32; ++j) sum += Edeprel[dr * 32 + j] * ew[j];
#pragma unroll 8
    for (int j = 0; j < 32; ++j) sum += Edeparc[da * 32 + j] * ew[32 + j];
    const float* ea = &expatt[((size_t)e * NH + h) * ATTD];
    const float* dn = &denom[(size_t)t * HKV + h * DV];
#pragma unroll 8
    for (int n = 0; n < 32; ++n) sum += (ea[n] / (dn[n] + EPSV)) * ew[64 + n];

    const float score = lrelu(sum);
    attn_out[(size_t)e * NH + h] = score;

    const float* vs = &v[(size_t)s * HKV + h * DV];
    float* ac       = &accum[(size_t)t * HKV + h * DV];
#pragma unroll 8
    for (int n = 0; n < 32; ++n) atomicAdd(&ac[n], vs[n] * score);
}

__global__ __launch_bounds__(256) void zero_kernel(float* __restrict__ buf,
                                                   unsigned* __restrict__ maxscr,
                                                   size_t n) {
    const size_t i = (size_t)blockIdx.x * 256 + threadIdx.x;
    if (i < n) buf[i] = 0.f;
    if (i == 0) *maxscr = 0u;   // decodes below any encoded real score
}

// ---------------------------------------------------------------------------
extern "C" void kernel_launch(void* const* d_in, const int* in_sizes, int n_in,
                              void* d_out, int out_size, void* d_ws, size_t ws_size,
                              hipStream_t stream) {
    const float* inp     = (const float*)d_in[0];
    const int*   relpos  = (const int*)d_in[1];
    // d_in[2] (word_rel_pos_edge) and d_in[3] (deprel_ext_edge) unused by ref
    const int*   eidx    = (const int*)d_in[4];
    const int*   src     = eidx;
    const int*   tgt     = eidx + N_EDGES;
    const int*   deprel  = (const int*)d_in[5];
    const int*   deparc  = (const int*)d_in[6];
    const float* Wkq     = (const float*)d_in[7];
    const float* Wv      = (const float*)d_in[8];
    const float* att_w   = (const float*)d_in[9];
    const float* edge_w  = (const float*)d_in[10];
    const float* Wf      = (const float*)d_in[11];
    const float* bias    = (const float*)d_in[12];
    const float* Edeprel = (const float*)d_in[13];
    const float* Edeparc = (const float*)d_in[14];
    const float* Epos    = (const float*)d_in[15];

    float* out      = (float*)d_out;
    float* attn_out = out + (size_t)N_NODES * DMODEL;

    // workspace carve-up (floats): kq | v | denom | accum | scores | maxscr
    float* W      = (float*)d_ws;
    float* kq     = W;
    float* v      = kq + (size_t)N_NODES * HKV;
    float* denom  = v + (size_t)N_NODES * HKV;
    float* accum  = denom + (size_t)N_NODES * HKV;
    float* scores = accum + (size_t)N_NODES * HKV;
    unsigned* maxscr = (unsigned*)(scores + (size_t)N_EDGES * HKV);

    // 1) zero denom + accum (contiguous) and max scratch
    const size_t zn = (size_t)2 * N_NODES * HKV;
    zero_kernel<<<(unsigned)((zn + 255) / 256), 256, 0, stream>>>(denom, maxscr, zn);

    // 2-3) kq = inp @ Wkq ; v = inp @ Wv   (M=50000, K=256, N=128)
    dim3 g1((MTILES + 7) / 8, HKV / 64);
    gemm_wmma<<<g1, 256, 0, stream>>>(inp, Wkq, nullptr, kq, N_NODES, HKV, DMODEL);
    gemm_wmma<<<g1, 256, 0, stream>>>(inp, Wv, nullptr, v, N_NODES, HKV, DMODEL);

    // 4) per-edge attention scores + global max
    dim3 g2((ETILES + 7) / 8, NH);
    edge_att_kernel<<<g2, 256, 0, stream>>>(kq, src, tgt, relpos, Epos, att_w,
                                            scores, maxscr);

    // 5) exp + denominator segment-sum
    const size_t en = (size_t)N_EDGES * HKV;
    exp_denom_kernel<<<(unsigned)((en + 255) / 256), 256, 0, stream>>>(
        scores, tgt, maxscr, denom);

    // 6) edge scores, attn output, weighted scatter into accum
    edge_final_kernel<<<(N_EDGES * NH + 255) / 256, 256, 0, stream>>>(
        scores, denom, src, tgt, deprel, deparc, Edeprel, Edeparc, edge_w, v,
        attn_out, accum);

    // 7) out = accum @ Wf + bias   (M=50000, K=128, N=256)
    dim3 g3((MTILES + 7) / 8, DMODEL / 64);
    gemm_wmma<<<g3, 256, 0, stream>>>(accum, Wf, bias, out, N_NODES, DMODEL, HKV);
}